// FeatureInteraction_5677946766094
// MI455X (gfx1250) — compile-verified
//
#include <hip/hip_runtime.h>

// CDNA5 / gfx1250 — wave32 WMMA.
// out[b, packed(i,j)] = x[b,i]*x[b,j], j>=i, packed row-major upper triangle.
// One wave = one (batch b, tile-row ti); walks tiles tj = ti..15.
// V_WMMA_F32_16X16X4_F32 per 16x16 tile (K=0 live, K=1..3 zero) -> exact f32.

typedef float v2f __attribute__((ext_vector_type(2)));
typedef float v8f __attribute__((ext_vector_type(8)));

#define DIM 256
#define NTILE 16           // DIM / 16
#define OUT_PER_B 32896    // DIM*(DIM+1)/2
#define WAVES_PER_BLOCK 8  // 256 threads / wave32

__global__ __launch_bounds__(256)
void feature_interaction_wmma(const float* __restrict__ x,
                              float* __restrict__ out) {
  const int lane = threadIdx.x & 31;
  const int l16  = lane & 15;
  const int half = lane >> 4;  // 0: lanes 0-15 (M=r), 1: lanes 16-31 (M=r+8)

  // wave id -> (b, ti); readfirstlane => SGPRs => all tile guards are scalar
  // branches (EXEC stays all-1s for every WMMA).
  const int ws = __builtin_amdgcn_readfirstlane(
      (int)blockIdx.x * WAVES_PER_BLOCK + ((int)threadIdx.x >> 5));
  const int b     = ws >> 4;
  const int ti    = ws & 15;
  const int bbase = b * OUT_PER_B;  // < 2^27, int-safe

  const float* xb = x + (size_t)b * DIM;

  // A (16x4 f32): VGPR0 lanes 0-15 = K=0 column = x[16*ti + m]; rest zero.
  const float xi = xb[ti * 16 + l16];
  v2f A;  A.x = half ? 0.0f : xi;  A.y = 0.0f;

  // Packed-index chain anchored at tj = 0:
  //   idxr[r] = bbase + rowoff(i0+r) + (l16 - (i0+r)),  rowoff(i)=i*256-i(i-1)/2
  // row-to-row delta: 255 - i;  tile-to-tile delta: +16 (compile-time const).
  const int i0 = ti * 16 + half * 8;
  int idxr[8];
  idxr[0] = bbase + i0 * 256 - ((i0 * (i0 - 1)) >> 1) + (l16 - i0);
  int d = 255 - i0;
#pragma unroll
  for (int r = 1; r < 8; ++r) { idxr[r] = idxr[r - 1] + d; d -= 1; }

  // ---- diagonal tile (tj == ti): B == A slice; symmetric duplicate-write ----
  {
    v8f C = {};
    C = __builtin_amdgcn_wmma_f32_16x16x4_f32(false, A, false, A,
                                              (short)0, C, false, false);
    const int jd = ti * 16 + l16;
#pragma unroll
    for (int r = 0; r < 8; ++r) {
      const int i  = i0 + r;
      const int a  = (i < jd) ? i : jd;   // min
      const int bx = (i < jd) ? jd : i;   // max; x_i*x_j == x_j*x_i bitwise
      const int off = bbase + a * 256 - ((a * (a - 1)) >> 1) + (bx - a);
      __builtin_nontemporal_store(C[r], out + off);
    }
  }

  // ---- off-diagonal tiles tj = ti+1 .. 15 (scalar skip; imm-offset stores) ----
#pragma unroll
  for (int tj = 1; tj < NTILE; ++tj) {
    if (tj > ti) {  // ti is SGPR, tj is literal -> s_cmp + s_cbranch
      const float xj = xb[tj * 16 + l16];
      v2f Bv;  Bv.x = half ? 0.0f : xj;  Bv.y = 0.0f;
      v8f C = {};
      C = __builtin_amdgcn_wmma_f32_16x16x4_f32(false, A, false, Bv,
                                                (short)0, C, false, false);
#pragma unroll
      for (int r = 0; r < 8; ++r) {
        __builtin_nontemporal_store(C[r], out + (idxr[r] + tj * 16));
      }
    }
  }
}

extern "C" void kernel_launch(void* const* d_in, const int* in_sizes, int n_in,
                              void* d_out, int out_size, void* d_ws, size_t ws_size,
                              hipStream_t stream) {
  const float* x  = (const float*)d_in[0];
  float*      out = (float*)d_out;

  const int B = in_sizes[0] / DIM;                 // 4096
  const int waves  = B * NTILE;                    // 65,536
  const int blocks = waves / WAVES_PER_BLOCK;      // 8,192 (B % 8 == 0)

  feature_interaction_wmma<<<blocks, 256, 0, stream>>>(x, out);
}